// DQGCN_7017976561678
// MI455X (gfx1250) — compile-verified
//
#include <hip/hip_runtime.h>
#include <math.h>

typedef float v2f __attribute__((ext_vector_type(2)));
typedef float v8f __attribute__((ext_vector_type(8)));

#define N_ENTS   50000
#define HDIM     128
#define NREL2    500
#define NEDGE    400000L
#define NHELEM   ((long)N_ENTS * HDIM)
#define RRELU_SLOPE 0.22916666666666666f

// ---------------------------------------------------------------------------
// Generic K-chunked WMMA f32 GEMM: C[M,128] = sum_s scale(A_s) @ B_s
// Block = 128 threads (4 waves), each wave owns a 16x128 output tile.
// B chunk staged transposed in LDS (stride 66 => bank-conflict free),
// A chunk staged per-wave with optional per-row scaling (norm / has_in masks).
// ---------------------------------------------------------------------------
struct GemmSrc { const float* A; const float* B; long lda; int K; int sm; };
struct GemmDesc {
  GemmSrc s[3];
  int nSrc;
  const float* rowscale;  // norm vector (sm: 1=*norm, 2=*has_in, 3=*!has_in)
  float* C;
  int M;
  int epi;                // 0=store, 1=l2norm rows, 2=rrelu
};

__global__ __launch_bounds__(128) void gemm_wmma_kernel(GemmDesc d) {
  __shared__ __attribute__((aligned(16))) float sBT[128 * 66];
  __shared__ __attribute__((aligned(16))) float sA[4][16 * 66];
  const int tid   = threadIdx.x;
  const int w     = tid >> 5;
  const int lane  = tid & 31;
  const int laneM = lane & 15;
  const int laneH = lane >> 4;
  const long rbase = (long)blockIdx.x * 64 + (long)w * 16;

  v8f acc[8] = {};

  for (int s = 0; s < d.nSrc; ++s) {
    const float* __restrict__ A = d.s[s].A;
    const float* __restrict__ B = d.s[s].B;
    const long lda = d.s[s].lda;
    const int  K   = d.s[s].K;
    const int  sm  = d.s[s].sm;
    for (int kb = 0; kb < K; kb += 64) {
      const int Kc = (K - kb < 64) ? (K - kb) : 64;   // always multiple of 4
      // stage B chunk transposed: sBT[n][k] = B[kb+k][n]
      for (int idx = tid; idx < 128 * Kc; idx += 128) {
        const int k = idx >> 7;
        const int n = idx & 127;
        sBT[n * 66 + k] = B[(long)(kb + k) * 128 + n];
      }
      // stage A rows for this wave, with per-row scale
      for (int idx = lane; idx < 16 * Kc; idx += 32) {
        const int r = idx / Kc;
        const int k = idx - r * Kc;
        const long row = rbase + r;
        float v = 0.0f;
        if (row < d.M) {
          v = A[row * lda + kb + k];
          if (sm) {
            const float nm = d.rowscale[row];
            if (sm == 1)      v *= nm;
            else if (sm == 2) v = (nm > 0.0f) ? v : 0.0f;
            else              v = (nm > 0.0f) ? 0.0f : v;
          }
        }
        sA[w][r * 66 + k] = v;
      }
      __syncthreads();
      const int nkk = Kc >> 2;
      for (int kk = 0; kk < nkk; ++kk) {
        const v2f a = *(const v2f*)&sA[w][laneM * 66 + (kk << 2) + (laneH << 1)];
#pragma unroll
        for (int j = 0; j < 8; ++j) {
          const v2f b = *(const v2f*)&sBT[(j * 16 + laneM) * 66 + (kk << 2) + (laneH << 1)];
          acc[j] = __builtin_amdgcn_wmma_f32_16x16x4_f32(
              false, a, false, b, (short)0, acc[j], false, false);
        }
      }
      __syncthreads();
    }
  }

  if (d.epi == 1) {  // per-row l2 normalize (row i -> rbase + i + 8*laneH)
    float ss[8];
#pragma unroll
    for (int i = 0; i < 8; ++i) ss[i] = 0.0f;
#pragma unroll
    for (int j = 0; j < 8; ++j)
#pragma unroll
      for (int i = 0; i < 8; ++i) ss[i] += acc[j][i] * acc[j][i];
#pragma unroll
    for (int m = 1; m < 16; m <<= 1)
#pragma unroll
      for (int i = 0; i < 8; ++i) ss[i] += __shfl_xor(ss[i], m, 32);
#pragma unroll
    for (int i = 0; i < 8; ++i) ss[i] = 1.0f / fmaxf(sqrtf(ss[i]), 1e-12f);
#pragma unroll
    for (int j = 0; j < 8; ++j)
#pragma unroll
      for (int i = 0; i < 8; ++i) acc[j][i] *= ss[i];
  } else if (d.epi == 2) {  // rrelu (eval-mode slope)
#pragma unroll
    for (int j = 0; j < 8; ++j)
#pragma unroll
      for (int i = 0; i < 8; ++i) {
        const float v = acc[j][i];
        acc[j][i] = (v >= 0.0f) ? v : v * RRELU_SLOPE;
      }
  }
#pragma unroll
  for (int j = 0; j < 8; ++j)
#pragma unroll
    for (int i = 0; i < 8; ++i) {
      const long row = rbase + i + (laneH ? 8 : 0);
      if (row < d.M) d.C[row * 128 + j * 16 + laneM] = acc[j][i];
    }
}

// ---------------------------------------------------------------------------
// Edge scatter: pre_agg[dst] += h[src] + emb_rel[etype]  (32 lanes per edge)
// ---------------------------------------------------------------------------
__global__ __launch_bounds__(256) void scatter_kernel(
    const float* __restrict__ h, const float* __restrict__ er,
    const int* __restrict__ src, const int* __restrict__ dst,
    const int* __restrict__ et, float* __restrict__ pre, long E) {
  const long t = (long)blockIdx.x * blockDim.x + threadIdx.x;
  if (t >= E * 32) return;
  const long e = t >> 5;
  const int  q = (int)(t & 31);
  const int  s = src[e];
  const int  d = dst[e];
  const int  r = et[e];
  const float4 hv = *(const float4*)(h  + (long)s * 128 + q * 4);
  const float4 rv = *(const float4*)(er + (long)r * 128 + q * 4);
  float* p = pre + (long)d * 128 + q * 4;
  atomicAdd(p + 0, hv.x + rv.x);
  atomicAdd(p + 1, hv.y + rv.y);
  atomicAdd(p + 2, hv.z + rv.z);
  atomicAdd(p + 3, hv.w + rv.w);
}

__global__ void deg_kernel(const int* __restrict__ dst, float* __restrict__ deg, long E) {
  const long e = (long)blockIdx.x * blockDim.x + threadIdx.x;
  if (e < E) atomicAdd(&deg[dst[e]], 1.0f);
}

__global__ void norm_kernel(float* __restrict__ deg, int N) {
  const int i = blockIdx.x * blockDim.x + threadIdx.x;
  if (i < N) { const float v = deg[i]; deg[i] = (v > 0.0f) ? (1.0f / v) : 0.0f; }
}

__global__ void zero_kernel(float* __restrict__ p, long n) {
  const long i = (long)blockIdx.x * blockDim.x + threadIdx.x;
  if (i < n) p[i] = 0.0f;
}

__global__ void copy_kernel(const float* __restrict__ a, float* __restrict__ b, long n) {
  const long i = (long)blockIdx.x * blockDim.x + threadIdx.x;
  if (i < n) b[i] = a[i];
}

__global__ void timevec_kernel(const float* __restrict__ a, const float* __restrict__ b,
                               const int* __restrict__ tptr, float* __restrict__ tv, long n) {
  const long i = (long)blockIdx.x * blockDim.x + threadIdx.x;
  if (i >= n) return;
  const float t = (float)tptr[0];
  tv[i] = 0.5f * a[i] * t + 0.5f * sinf(6.283185307179586f * b[i] * t);
}

// ---------------------------------------------------------------------------
// Fused time-gate: per row n
//   cur = l2norm(h[n]); tw_i = (in_i[n]+related[n]) @ tg_w + tg_b
//   softmax over {tw_0..tw_{g}, 0}; out = l2norm(cur*w_last + sum w_i*in_i)
// ---------------------------------------------------------------------------
__global__ __launch_bounds__(128) void combine_kernel(
    const float* __restrict__ hfin, const float* __restrict__ related,
    const float* __restrict__ in0, const float* __restrict__ in1,
    const float* __restrict__ in2, int nIn,
    const float* __restrict__ tgw, const float* __restrict__ tgb,
    float* __restrict__ outp, int M) {
  __shared__ float s_in[3][128];
  __shared__ float s_rel[128];
  __shared__ float s_h[128];
  __shared__ float s_red[128];
  const int c = threadIdx.x;
  const float* ins[3] = { in0, in1, in2 };
  const float bias = tgb[c];
  for (long n = blockIdx.x; n < M; n += gridDim.x) {
    const long base = n * 128;
    s_rel[c] = related[base + c];
    s_h[c]   = hfin[base + c];
    for (int i = 0; i < nIn; ++i) s_in[i][c] = ins[i][base + c];
    __syncthreads();
    s_red[c] = s_h[c] * s_h[c];
    __syncthreads();
    for (int off = 64; off > 0; off >>= 1) {
      if (c < off) s_red[c] += s_red[c + off];
      __syncthreads();
    }
    const float cur = s_h[c] / fmaxf(sqrtf(s_red[0]), 1e-12f);
    __syncthreads();
    float tv[3];
    for (int i = 0; i < nIn; ++i) {
      float a = bias;
      for (int k = 0; k < 128; ++k) a += (s_in[i][k] + s_rel[k]) * tgw[k * 128 + c];
      tv[i] = a;
    }
    float mx = 0.0f;  // implicit zeros row
    for (int i = 0; i < nIn; ++i) mx = fmaxf(mx, tv[i]);
    const float ez = expf(-mx);
    float den = ez;
    float e[3];
    for (int i = 0; i < nIn; ++i) { e[i] = expf(tv[i] - mx); den += e[i]; }
    float o = cur * (ez / den);
    for (int i = 0; i < nIn; ++i) o += (e[i] / den) * s_in[i][c];
    s_red[c] = o * o;
    __syncthreads();
    for (int off = 64; off > 0; off >>= 1) {
      if (c < off) s_red[c] += s_red[c + off];
      __syncthreads();
    }
    outp[base + c] = o / fmaxf(sqrtf(s_red[0]), 1e-12f);
    __syncthreads();
  }
}

// ---------------------------------------------------------------------------
extern "C" void kernel_launch(void* const* d_in, const int* in_sizes, int n_in,
                              void* d_out, int out_size, void* d_ws, size_t ws_size,
                              hipStream_t stream) {
  const float* err_mat    = (const float*)d_in[0];
  const float* emb_rel    = (const float*)d_in[1];
  const float* static_emb = (const float*)d_in[2];
  const float* alpha_t    = (const float*)d_in[3];
  const float* beta_t     = (const float*)d_in[4];
  const float* temporal_w = (const float*)d_in[5];
  const float* tg_w       = (const float*)d_in[6];
  const float* tg_b       = (const float*)d_in[7];
  const float* w_neigh    = (const float*)d_in[8];
  const float* w_loop     = (const float*)d_in[9];
  const float* w_evolve   = (const float*)d_in[10];
  const int*   src        = (const int*)d_in[11];
  const int*   dst        = (const int*)d_in[12];
  const int*   etype      = (const int*)d_in[13];
  const int*   tptr       = (const int*)d_in[14];
  float* out = (float*)d_out;

  float* f = (float*)d_ws;
  float* related  = f;
  float* inbuf[3] = { f + NHELEM, f + 2 * NHELEM, f + 3 * NHELEM };
  float* hA       = f + 4 * NHELEM;
  float* hB       = f + 5 * NHELEM;
  float* pre      = f + 6 * NHELEM;
  float* normv    = f + 7 * NHELEM;

  const int gemmGrid = (N_ENTS + 63) / 64;

  // 1) related = err_mat @ emb_rel   (K=500 -> chunks 64..64,52)
  {
    GemmDesc d = {};
    d.s[0] = GemmSrc{ err_mat, emb_rel, (long)NREL2, NREL2, 0 };
    d.nSrc = 1; d.rowscale = normv; d.C = related; d.M = N_ENTS; d.epi = 0;
    hipLaunchKernelGGL(gemm_wmma_kernel, dim3(gemmGrid), dim3(128), 0, stream, d);
  }
  // 2) timevec (into hA)
  hipLaunchKernelGGL(timevec_kernel, dim3((unsigned)((NHELEM + 255) / 256)), dim3(256),
                     0, stream, alpha_t, beta_t, tptr, hA, NHELEM);
  // 3) x0 = l2norm([static_emb, timevec] @ temporal_w)  (two K=128 sources)
  {
    GemmDesc d = {};
    d.s[0] = GemmSrc{ static_emb, temporal_w,             128L, 128, 0 };
    d.s[1] = GemmSrc{ hA,         temporal_w + 128 * 128, 128L, 128, 0 };
    d.nSrc = 2; d.rowscale = normv; d.C = inbuf[0]; d.M = N_ENTS; d.epi = 1;
    hipLaunchKernelGGL(gemm_wmma_kernel, dim3(gemmGrid), dim3(128), 0, stream, d);
  }

  for (int g = 0; g < 3; ++g) {
    const int* sg = src   + (long)g * NEDGE;
    const int* dg = dst   + (long)g * NEDGE;
    const int* eg = etype + (long)g * NEDGE;

    hipLaunchKernelGGL(zero_kernel, dim3((N_ENTS + 255) / 256), dim3(256), 0, stream,
                       normv, (long)N_ENTS);
    hipLaunchKernelGGL(deg_kernel, dim3((unsigned)((NEDGE + 255) / 256)), dim3(256),
                       0, stream, dg, normv, NEDGE);
    hipLaunchKernelGGL(norm_kernel, dim3((N_ENTS + 255) / 256), dim3(256), 0, stream,
                       normv, N_ENTS);

    const float* h = inbuf[g];
    float* lay[2] = { hA, hB };
    for (int l = 0; l < 2; ++l) {
      hipLaunchKernelGGL(zero_kernel, dim3((unsigned)((NHELEM + 255) / 256)), dim3(256),
                         0, stream, pre, NHELEM);
      hipLaunchKernelGGL(scatter_kernel, dim3((unsigned)((NEDGE * 32 + 255) / 256)),
                         dim3(256), 0, stream, h, emb_rel, sg, dg, eg, pre, NEDGE);
      // h_new = rrelu(norm*pre @ Wn + has_in*h @ Wl + !has_in*h @ We)  (K=384)
      GemmDesc d = {};
      const long wo = (long)l * 128 * 128;
      d.s[0] = GemmSrc{ pre, w_neigh  + wo, 128L, 128, 1 };
      d.s[1] = GemmSrc{ h,   w_loop   + wo, 128L, 128, 2 };
      d.s[2] = GemmSrc{ h,   w_evolve + wo, 128L, 128, 3 };
      d.nSrc = 3; d.rowscale = normv; d.C = lay[l]; d.M = N_ENTS; d.epi = 2;
      hipLaunchKernelGGL(gemm_wmma_kernel, dim3(gemmGrid), dim3(128), 0, stream, d);
      h = lay[l];
    }

    float* og = (g < 2) ? inbuf[g + 1] : out;
    hipLaunchKernelGGL(combine_kernel, dim3(6250), dim3(128), 0, stream,
                       hB, related, inbuf[0], inbuf[1], inbuf[2], g + 1,
                       tg_w, tg_b, og, N_ENTS);
  }

  // tuple output: append emb_rel after the final embedding
  hipLaunchKernelGGL(copy_kernel, dim3(((long)NREL2 * HDIM + 255) / 256), dim3(256),
                     0, stream, emb_rel, out + NHELEM, (long)NREL2 * HDIM);
}